// SpectralActivation_46712064312051
// MI455X (gfx1250) — compile-verified
//
#include <hip/hip_runtime.h>
#include <cstdint>

// SpectralActivation(x): SVD of the 4x4 circulant built from each 4-vector,
// relu on singular values (already >= 0 -> exact no-op), reconstruct, take
// first column == the original 4-vector. The whole op is the identity map.
//
// => Optimal MI455X kernel: pure HBM-bandwidth streaming copy.
//    32 MiB in + 32 MiB out = 67 MB -> ~2.9 us @ 23.3 TB/s.
//
// Implemented on the CDNA5 async global<->LDS data movers (gfx1250-only):
//   global_load_async_to_lds_b128  ->  s_wait_asynccnt  ->
//   global_store_async_from_lds_b128
// Data never touches VGPRs. Double-buffered LDS staging with ONE full
// ASYNCcnt drain per batch (loads/stores retire out of order across types,
// so partial waits cannot distinguish them; wait-0 per batch is the safe
// minimum). Final store batch drains via S_ENDPGM's implicit wait-idle.
// Full batches take an unguarded path (uniform scalar branch) so the hot
// loop has zero exec-mask churn around the async ops.

#define TPB   256                 // 8 wave32s per workgroup
#define SLOTS 8                   // 16B tiles / thread / batch -> 4 KB in flight per wave
#define BATCH (TPB * SLOTS)       // 2048 tiles = 32 KiB per LDS buffer

template <bool GUARD>
__device__ __forceinline__ void issue_async_loads(
    const float* src, const float4* buf, unsigned base, unsigned ntiles, unsigned tid)
{
#pragma unroll
    for (int s = 0; s < SLOTS; ++s) {
        unsigned tile = base + (unsigned)s * TPB + tid;      // coalesced per slot
        if (!GUARD || tile < ntiles) {
            // Low 32 bits of a generic LDS pointer == LDS byte offset (ISA 10.2).
            uint32_t loff = (uint32_t)(uintptr_t)(const void*)&buf[s * TPB + tid];
            uint32_t goff = tile * 16u;                      // byte offset (<32 MiB)
            asm volatile("global_load_async_to_lds_b128 %0, %1, %2"
                         :
                         : "v"(loff), "v"(goff), "s"(src)
                         : "memory");
        }
    }
}

template <bool GUARD>
__device__ __forceinline__ void issue_async_stores(
    float* dst, const float4* buf, unsigned base, unsigned ntiles, unsigned tid)
{
#pragma unroll
    for (int s = 0; s < SLOTS; ++s) {
        unsigned tile = base + (unsigned)s * TPB + tid;
        if (!GUARD || tile < ntiles) {
            uint32_t loff = (uint32_t)(uintptr_t)(const void*)&buf[s * TPB + tid];
            uint32_t goff = tile * 16u;
            asm volatile("global_store_async_from_lds_b128 %0, %1, %2"
                         :
                         : "v"(goff), "v"(loff), "s"(dst)
                         : "memory");
        }
    }
}

__global__ __launch_bounds__(TPB) void spectral_identity_copy_async(
    const float* __restrict__ src, float* __restrict__ dst,
    unsigned ntiles, unsigned nelems)
{
    __shared__ float4 smem[2][BATCH];                        // 64 KiB of 320 KiB/WGP
    const unsigned tid    = threadIdx.x;
    const unsigned stride = gridDim.x * BATCH;               // tiles per grid pass

    unsigned cur = blockIdx.x * BATCH;
    if (cur < ntiles) {
        unsigned buf = 0;
        // Prologue loads (uniform full/ragged selection -> scalar branch).
        if (cur + BATCH <= ntiles) issue_async_loads<false>(src, smem[0], cur, ntiles, tid);
        else                       issue_async_loads<true >(src, smem[0], cur, ntiles, tid);
        for (;;) {
            unsigned next = cur + stride;                    // uniform -> scalar branch
            // Drains loads(cur, buf) AND any prior stores on buf^1:
            asm volatile("s_wait_asynccnt 0x0" ::: "memory");
            if (next < ntiles) {
                if (next + BATCH <= ntiles)
                    issue_async_loads<false>(src, smem[buf ^ 1u], next, ntiles, tid);
                else
                    issue_async_loads<true >(src, smem[buf ^ 1u], next, ntiles, tid);
            }
            if (cur + BATCH <= ntiles)
                issue_async_stores<false>(dst, smem[buf], cur, ntiles, tid);
            else
                issue_async_stores<true >(dst, smem[buf], cur, ntiles, tid);
            if (next >= ntiles) break;                       // stores drain at S_ENDPGM
            cur = next;
            buf ^= 1u;
        }
    }

    // Scalar tail for nelems % 4 (not hit for 8192x1024; kept for robustness).
    unsigned tailBase = ntiles * 4u;
    if (blockIdx.x == 0 && tid < (nelems - tailBase)) {
        dst[tailBase + tid] = src[tailBase + tid];
    }
}

extern "C" void kernel_launch(void* const* d_in, const int* in_sizes, int n_in,
                              void* d_out, int out_size, void* d_ws, size_t ws_size,
                              hipStream_t stream) {
    (void)in_sizes; (void)n_in; (void)d_ws; (void)ws_size;
    const float* x   = (const float*)d_in[0];
    float*       out = (float*)d_out;

    const unsigned n      = (unsigned)out_size;   // 8 388 608 f32 elements
    const unsigned ntiles = n / 4u;               // 2 097 152 x 16B tiles

    unsigned blocks = (ntiles + BATCH - 1u) / BATCH;
    if (blocks > 1024u) blocks = 1024u;           // 1024 * 2048 tiles = exactly ntiles
    if (blocks == 0u) blocks = 1u;

    spectral_identity_copy_async<<<blocks, TPB, 0, stream>>>(x, out, ntiles, n);
}